// CenterNet_39049842655966
// MI455X (gfx1250) — compile-verified
//
#include <hip/hip_runtime.h>
#include <stdint.h>

typedef unsigned int u32;
typedef uint64_t u64;
typedef u32 u32x4 __attribute__((ext_vector_type(4)));
typedef int i32x4 __attribute__((ext_vector_type(4)));
typedef int i32x8 __attribute__((ext_vector_type(8)));

#define HW 128
#define IMG_ELEMS (HW * HW)

// One workgroup per 128x128 image. TDM pulls the whole 64KB image into LDS in
// one descriptor; 256 threads then slide a vertical 3-row window down their
// column, doing 3 LDS loads + ~8 v_max per output, and write coalesced.
__global__ __launch_bounds__(256)
void nms3x3_tdm_kernel(const float* __restrict__ in, float* __restrict__ out) {
    __shared__ float tile[IMG_ELEMS];

    const int img = (int)blockIdx.x;
    const u64 gaddr = (u64)(uintptr_t)(in + (size_t)img * IMG_ELEMS);

    // ---- Tensor Data Mover: global (row-major 128x128 f32) -> LDS ----
    if (threadIdx.x < 32) {  // one wave issues the DMA (uniform branch)
        const u32 lds_addr = (u32)(uintptr_t)(&tile[0]);  // low 32b = LDS byte offset

        u32x4 g0;
        g0.x = 1u;                                    // count=1, user desc, no gather
        g0.y = lds_addr;                              // bits 63:32  lds_addr
        g0.z = (u32)(gaddr & 0xFFFFFFFFu);            // bits 95:64  global_addr lo
        g0.w = (u32)((gaddr >> 32) & 0x01FFFFFFu)     // bits 120:96 global_addr hi
             | (2u << 30);                            // bits 127:126 type=2 (image)

        i32x8 g1;
        g1[0] = (int)(2u << 16);       // workgroup_mask=0, data_size=2 (4 bytes)
        g1[1] = (int)(128u << 16);     // [15:0] atomic_bar_addr=0, [31:16] tensor_dim0.lo = 128
        g1[2] = (int)(128u << 16);     // [15:0] tensor_dim0.hi=0, [31:16] tensor_dim1.lo = 128
        g1[3] = (int)(128u << 16);     // [15:0] tensor_dim1.hi=0, [31:16] tile_dim0 = 128
        g1[4] = (int)(128u);           // [15:0] tile_dim1 = 128,  [31:16] tile_dim2 = 0
        g1[5] = (int)(128u);           // tensor_dim0_stride.lo = 128 elements
        g1[6] = 0;                     // tensor_dim0_stride.hi, tensor_dim1_stride.lo
        g1[7] = 0;                     // tensor_dim1_stride.hi

        i32x4 z4 = {0, 0, 0, 0};       // groups 2/3 unused (2-D tensor)
#if defined(__clang_major__) && (__clang_major__ >= 23)
        i32x8 z8 = {0, 0, 0, 0, 0, 0, 0, 0};
        __builtin_amdgcn_tensor_load_to_lds(g0, g1, z4, z4, z8, 0);
#else
        __builtin_amdgcn_tensor_load_to_lds(g0, g1, z4, z4, 0);
#endif
        __builtin_amdgcn_s_wait_tensorcnt(0);
    }
    __syncthreads();

    // ---- 3x3 NMS with zero padding, vertical sliding window ----
    const int t  = (int)threadIdx.x;
    const int c  = t & (HW - 1);   // column 0..127
    const int hh = t >> 7;         // which half of the rows
    const int r0 = hh << 6;        // 0 or 64
    const bool cl = (c > 0);
    const bool cr = (c < HW - 1);

    float a0, a1, a2;   // row r-1 at cols c-1,c,c+1 (0 outside image)
    float b0, b1, b2;   // row r
    if (r0 == 0) {
        a0 = a1 = a2 = 0.0f;
    } else {
        const int row = (r0 - 1) * HW;
        a1 = tile[row + c];
        a0 = cl ? tile[row + c - 1] : 0.0f;
        a2 = cr ? tile[row + c + 1] : 0.0f;
    }
    {
        const int row = r0 * HW;
        b1 = tile[row + c];
        b0 = cl ? tile[row + c - 1] : 0.0f;
        b2 = cr ? tile[row + c + 1] : 0.0f;
    }

    float* __restrict__ outp = out + (size_t)img * IMG_ELEMS;

#pragma unroll 4
    for (int r = r0; r < r0 + 64; ++r) {
        float d0, d1, d2;   // row r+1
        if (r + 1 < HW) {
            const int row = (r + 1) * HW;
            d1 = tile[row + c];
            d0 = cl ? tile[row + c - 1] : 0.0f;
            d2 = cr ? tile[row + c + 1] : 0.0f;
        } else {
            d0 = d1 = d2 = 0.0f;
        }
        const float m = fmaxf(fmaxf(fmaxf(a0, a1), fmaxf(a2, b0)),
                              fmaxf(fmaxf(b1, b2), fmaxf(fmaxf(d0, d1), d2)));
        outp[r * HW + c] = (b1 == m) ? b1 : 0.0f;

        a0 = b0; a1 = b1; a2 = b2;
        b0 = d0; b1 = d1; b2 = d2;
    }
}

extern "C" void kernel_launch(void* const* d_in, const int* in_sizes, int n_in,
                              void* d_out, int out_size, void* d_ws, size_t ws_size,
                              hipStream_t stream) {
    const float* in = (const float*)d_in[0];
    float* out = (float*)d_out;
    const int n_imgs = in_sizes[0] / IMG_ELEMS;  // 16*80 = 1280
    nms3x3_tdm_kernel<<<n_imgs, 256, 0, stream>>>(in, out);
}